// HPR_WN_Top_K_72353019068522
// MI455X (gfx1250) — compile-verified
//
#include <hip/hip_runtime.h>
#include <hip/hip_bf16.h>

#define K_CLS 5
#define N_SUP 5
#define D_DIM 1024
#define Q_NUM 75000
#define N_AUG 70
#define TOPK  50
#define PADC  16
#define NPOOL (N_SUP + N_AUG)   // 75

typedef __attribute__((ext_vector_type(2))) float v2f;
typedef __attribute__((ext_vector_type(4))) float v4f;
typedef __attribute__((ext_vector_type(8))) float v8f;

// ---- workspace layout (float offsets) ----
constexpr int OFF_PROTO  = 0;                          // padded proto_e [16][1024]
constexpr int OFF_P2     = OFF_PROTO + PADC * D_DIM;   // ||proto||^2 [16]
constexpr int OFF_SUMS   = OFF_P2 + PADC;              // support sums [5][1024]
constexpr int OFF_SQS    = OFF_SUMS + K_CLS * D_DIM;   // support sq sums [5][1024]
constexpr int OFF_MEAN   = OFF_SQS + K_CLS * D_DIM;    // mean_c [5][1024]
constexpr int OFF_STD    = OFF_MEAN + K_CLS * D_DIM;   // std_c [5][1024]
constexpr int OFF_ZERO   = OFF_STD + K_CLS * D_DIM;    // ---- zeroed each launch ----
constexpr int OFF_SUMQ   = OFF_ZERO;                   // query sums [5][1024]
constexpr int OFF_SQQ    = OFF_SUMQ + K_CLS * D_DIM;   // query sq sums [5][1024]
constexpr int OFF_CNT    = OFF_SQQ + K_CLS * D_DIM;    // class counts [8]
constexpr int OFF_LOSS   = OFF_CNT + 8;                // loss accumulator [8]
constexpr int OFF_PROTON = OFF_LOSS + 8;               // padded proto_new [16][1024]
constexpr int OFF_P2N    = OFF_PROTON + PADC * D_DIM;  // ||proto_new||^2 [16]
constexpr int WS_END     = OFF_P2N + PADC;
constexpr int ZERO_CNT   = WS_END - OFF_ZERO;

// ------------------------------------------------------------------
// WMMA f32 16x16x4 dot-product tile over K=1024.
// aPtr: query row + 2*half ; bPtr: padded proto row (lane&15) + 2*half.
// Also accumulates this lane's partial sum-of-squares of the A row.
// ------------------------------------------------------------------
__device__ __forceinline__ v8f wmma_dot_rows(const float* __restrict__ aPtr,
                                             const float* __restrict__ bPtr,
                                             float& q2part) {
  v8f acc = {0.f, 0.f, 0.f, 0.f, 0.f, 0.f, 0.f, 0.f};
  float q2 = 0.f;
#pragma unroll 8
  for (int k0 = 0; k0 < D_DIM; k0 += 4) {
    v2f a = *(const v2f*)(aPtr + k0);
    v2f b = *(const v2f*)(bPtr + k0);
    q2 += a.x * a.x + a.y * a.y;
    acc = __builtin_amdgcn_wmma_f32_16x16x4_f32(false, a, false, b,
                                                (short)0, acc, false, false);
  }
  q2part = q2;
  return acc;
}

// ------------------------------------------------------------------
// threefry2x32-20 (JAX default PRNG) + inverse-erf normal
// ------------------------------------------------------------------
__device__ __forceinline__ void threefry2x32(unsigned k0, unsigned k1,
                                             unsigned x0, unsigned x1,
                                             unsigned& o0, unsigned& o1) {
  const unsigned R[8] = {13u, 15u, 26u, 6u, 17u, 29u, 16u, 24u};
  unsigned key[3] = {k0, k1, 0x1BD11BDAu ^ k0 ^ k1};
  x0 += k0; x1 += k1;
#pragma unroll
  for (int blk = 0; blk < 5; blk++) {
#pragma unroll
    for (int r = 0; r < 4; r++) {
      x0 += x1;
      unsigned rot = R[(blk & 1) * 4 + r];
      x1 = (x1 << rot) | (x1 >> (32 - rot));
      x1 ^= x0;
    }
    x0 += key[(blk + 1) % 3];
    x1 += key[(blk + 2) % 3] + (unsigned)(blk + 1);
  }
  o0 = x0; o1 = x1;
}

__device__ __forceinline__ float jax_normal(unsigned idx) {
  const unsigned total = (unsigned)K_CLS * N_AUG * D_DIM;   // 358400 (even)
  const unsigned halfn = total / 2u;
  unsigned c0, c1; bool lo;
  if (idx < halfn) { c0 = idx; c1 = idx + halfn; lo = true; }
  else             { c0 = idx - halfn; c1 = idx; lo = false; }
  unsigned o0, o1;
  threefry2x32(0u, 42u, c0, c1, o0, o1);
  unsigned bits = lo ? o0 : o1;
  float f = __uint_as_float((bits >> 9) | 0x3F800000u);     // [1,2)
  float u = 2.f * (f - 1.f) - 1.f;                          // [-1,1)
  u = fminf(fmaxf(u, -0.9999999f), 0.9999999f);
  return 1.41421356237f * erfinvf(u);
}

__device__ __forceinline__ float pool_val(const float* __restrict__ feat,
                                          const float* __restrict__ ws,
                                          int c, int j, int d) {
  if (j < N_SUP) return feat[(size_t)(c * N_SUP + j) * D_DIM + d];
  unsigned idx = ((unsigned)c * N_AUG + (unsigned)(j - N_SUP)) * D_DIM + (unsigned)d;
  return ws[OFF_MEAN + c * D_DIM + d] + ws[OFF_STD + c * D_DIM + d] * jax_normal(idx);
}

// ------------------------------------------------------------------
// Kernel 1: support stats + padded proto_e + ||proto||^2
// ------------------------------------------------------------------
__global__ void k_prep(const float* __restrict__ feat, float* __restrict__ ws) {
  int r = blockIdx.x;            // 0..15
  int tid = threadIdx.x;         // 256
  float p2part = 0.f;
  for (int d = tid; d < D_DIM; d += 256) {
    float s = 0.f, ss = 0.f;
    if (r < K_CLS) {
      for (int n = 0; n < N_SUP; n++) {
        float v = feat[(size_t)(r * N_SUP + n) * D_DIM + d];
        s += v; ss += v * v;
      }
      ws[OFF_SUMS + r * D_DIM + d] = s;
      ws[OFF_SQS  + r * D_DIM + d] = ss;
    }
    float p = s * (1.f / N_SUP);
    ws[OFF_PROTO + r * D_DIM + d] = p;
    p2part += p * p;
  }
  __shared__ float red[256];
  red[tid] = p2part; __syncthreads();
  for (int s = 128; s > 0; s >>= 1) { if (tid < s) red[tid] += red[tid + s]; __syncthreads(); }
  if (tid == 0) ws[OFF_P2 + r] = red[0];
}

// ------------------------------------------------------------------
// Kernel 2: main streaming pass — WMMA dists, argmin, loss,
// per-class sum/sq accumulation staged in LDS.
// Block = 256 (8 waves), each wave handles a 16-query tile.
// ------------------------------------------------------------------
__global__ void __launch_bounds__(256) k_pass1(const float* __restrict__ feat,
                                               const int* __restrict__ label,
                                               float* __restrict__ ws) {
  const float* queries = feat + (size_t)(K_CLS * N_SUP) * D_DIM;
  __shared__ float s_sumQ[K_CLS * D_DIM];
  __shared__ float s_sqQ[K_CLS * D_DIM];
  __shared__ float s_dots[8][16][16];
  __shared__ int   s_pred[8][16];
  __shared__ float s_cnt[K_CLS];
  __shared__ float s_loss;

  int tid = threadIdx.x;
  for (int i = tid; i < K_CLS * D_DIM; i += 256) { s_sumQ[i] = 0.f; s_sqQ[i] = 0.f; }
  if (tid < K_CLS) s_cnt[tid] = 0.f;
  if (tid == 0) s_loss = 0.f;
  __syncthreads();

  int wave = tid >> 5, lane = tid & 31;
  int half = lane >> 4, nrow = lane & 15;
  int qBase = (blockIdx.x * 8 + wave) * 16;
  int qclamp = min(qBase + nrow, Q_NUM - 1);

  const float* aPtr = queries + (size_t)qclamp * D_DIM + 2 * half;
  const float* bPtr = ws + OFF_PROTO + nrow * D_DIM + 2 * half;
  float q2h;
  v8f acc = wmma_dot_rows(aPtr, bPtr, q2h);
  float q2 = q2h + __shfl_xor(q2h, 16, 32);   // full ||q||^2 on lanes 0..15

#pragma unroll
  for (int r = 0; r < 8; r++) s_dots[wave][r + 8 * half][nrow] = acc[r];
  __syncthreads();

  if (lane < 16) {
    int q = qBase + lane;
    bool valid = q < Q_NUM;
    float dist[K_CLS];
    float best = 3.4e38f;
    int pred = 0;
#pragma unroll
    for (int c = 0; c < K_CLS; c++) {
      float dot = s_dots[wave][lane][c];
      float p2  = ws[OFF_P2 + c];
      float dd  = sqrtf(fmaxf(q2 + p2 - 2.f * dot, 1e-12f));
      dist[c] = dd;
      if (dd < best) { best = dd; pred = c; }
    }
    s_pred[wave][lane] = pred;
    if (valid) {
      atomicAdd(&s_cnt[pred], 1.f);
      // loss = -logits2[tgt], logits2 = log_softmax(log_softmax(-dist))
      float mx = -dist[0];
#pragma unroll
      for (int c = 1; c < K_CLS; c++) mx = fmaxf(mx, -dist[c]);
      float se = 0.f;
#pragma unroll
      for (int c = 0; c < K_CLS; c++) se += expf(-dist[c] - mx);
      float lse1 = mx + logf(se);
      float l1[K_CLS];
      float mx2 = -3.4e38f;
#pragma unroll
      for (int c = 0; c < K_CLS; c++) { l1[c] = -dist[c] - lse1; mx2 = fmaxf(mx2, l1[c]); }
      float se2 = 0.f;
#pragma unroll
      for (int c = 0; c < K_CLS; c++) se2 += expf(l1[c] - mx2);
      float lse2 = mx2 + logf(se2);
      int tgt = label[Q_NUM + q];
      atomicAdd(&s_loss, -(l1[tgt] - lse2));
    }
  }
  __syncthreads();

  // pass 2: accumulate each query row into its predicted class (LDS atomics)
  for (int mm = 0; mm < 16; mm++) {
    int q = qBase + mm;
    if (q >= Q_NUM) break;                    // wave-uniform
    int c = s_pred[wave][mm];
    const float* row = queries + (size_t)q * D_DIM;
#pragma unroll
    for (int it = 0; it < D_DIM / 128; it++) {
      int d = it * 128 + lane * 4;
      v4f v = *(const v4f*)(row + d);
      atomicAdd(&s_sumQ[c * D_DIM + d + 0], v.x);
      atomicAdd(&s_sumQ[c * D_DIM + d + 1], v.y);
      atomicAdd(&s_sumQ[c * D_DIM + d + 2], v.z);
      atomicAdd(&s_sumQ[c * D_DIM + d + 3], v.w);
      atomicAdd(&s_sqQ[c * D_DIM + d + 0], v.x * v.x);
      atomicAdd(&s_sqQ[c * D_DIM + d + 1], v.y * v.y);
      atomicAdd(&s_sqQ[c * D_DIM + d + 2], v.z * v.z);
      atomicAdd(&s_sqQ[c * D_DIM + d + 3], v.w * v.w);
    }
  }
  __syncthreads();

  for (int i = tid; i < K_CLS * D_DIM; i += 256) {
    atomicAdd(&ws[OFF_SUMQ + i], s_sumQ[i]);
    atomicAdd(&ws[OFF_SQQ + i], s_sqQ[i]);
  }
  if (tid < K_CLS) atomicAdd(&ws[OFF_CNT + tid], s_cnt[tid]);
  if (tid == 0) atomicAdd(&ws[OFF_LOSS], s_loss);
}

// ------------------------------------------------------------------
// Kernel 3: per-class mean/std
// ------------------------------------------------------------------
__global__ void k_stats(float* __restrict__ ws) {
  int c = blockIdx.x, tid = threadIdx.x;
  float m = (float)N_SUP + ws[OFF_CNT + c];
  for (int d = tid; d < D_DIM; d += 256) {
    float sum  = ws[OFF_SUMS + c * D_DIM + d] + ws[OFF_SUMQ + c * D_DIM + d];
    float mean = sum / m;
    float sq   = ws[OFF_SQS + c * D_DIM + d] + ws[OFF_SQQ + c * D_DIM + d];
    float var  = (sq - m * mean * mean) / (m - 1.f);
    ws[OFF_MEAN + c * D_DIM + d] = mean;
    ws[OFF_STD  + c * D_DIM + d] = sqrtf(fmaxf(var, 0.f));
  }
}

// ------------------------------------------------------------------
// Kernel 4: candidate distances, top-50 selection by rank, proto_new
// ------------------------------------------------------------------
__global__ void k_select(const float* __restrict__ feat, float* __restrict__ ws) {
  int c = blockIdx.x, tid = threadIdx.x, wave = tid >> 5, lane = tid & 31;
  __shared__ float d2p[80];
  __shared__ int   sel[80];

  for (int j = wave; j < NPOOL; j += 8) {
    float part = 0.f;
    for (int d = lane; d < D_DIM; d += 32) {
      float diff = pool_val(feat, ws, c, j, d) - ws[OFF_PROTO + c * D_DIM + d];
      part += diff * diff;
    }
    for (int off = 16; off > 0; off >>= 1) part += __shfl_down(part, off, 32);
    if (lane == 0) d2p[j] = sqrtf(part);
  }
  __syncthreads();

  if (tid < NPOOL) {
    float v = d2p[tid];
    int rank = 0;
    for (int i = 0; i < NPOOL; i++) {
      float vi = d2p[i];
      rank += (vi < v) || (vi == v && i < tid);   // ties -> lower index wins
    }
    sel[tid] = (rank < TOPK) ? 1 : 0;
  }
  __syncthreads();

  float p2part = 0.f;
  for (int d = tid; d < D_DIM; d += 256) {
    float s = 0.f;
    for (int j = 0; j < NPOOL; j++)
      if (sel[j]) s += pool_val(feat, ws, c, j, d);
    float pn = s * (1.f / TOPK);
    ws[OFF_PROTON + c * D_DIM + d] = pn;
    p2part += pn * pn;
  }
  __shared__ float red[256];
  red[tid] = p2part; __syncthreads();
  for (int s = 128; s > 0; s >>= 1) { if (tid < s) red[tid] += red[tid + s]; __syncthreads(); }
  if (tid == 0) ws[OFF_P2N + c] = red[0];
}

// ------------------------------------------------------------------
// Kernel 5: second streaming pass — dist_new via WMMA, softmax, output
// ------------------------------------------------------------------
__global__ void __launch_bounds__(256) k_pass2(const float* __restrict__ feat,
                                               const float* __restrict__ ws,
                                               float* __restrict__ out) {
  const float* queries = feat + (size_t)(K_CLS * N_SUP) * D_DIM;
  __shared__ float s_dots[8][16][16];
  int tid = threadIdx.x, wave = tid >> 5, lane = tid & 31;
  int half = lane >> 4, nrow = lane & 15;
  int qBase = (blockIdx.x * 8 + wave) * 16;
  int qclamp = min(qBase + nrow, Q_NUM - 1);

  const float* aPtr = queries + (size_t)qclamp * D_DIM + 2 * half;
  const float* bPtr = ws + OFF_PROTON + nrow * D_DIM + 2 * half;
  float q2h;
  v8f acc = wmma_dot_rows(aPtr, bPtr, q2h);
  float q2 = q2h + __shfl_xor(q2h, 16, 32);

#pragma unroll
  for (int r = 0; r < 8; r++) s_dots[wave][r + 8 * half][nrow] = acc[r];
  __syncthreads();

  if (lane < 16 && (qBase + lane) < Q_NUM) {
    int q = qBase + lane;
    float z[K_CLS];
    float mx = -3.4e38f;
#pragma unroll
    for (int c = 0; c < K_CLS; c++) {
      float dot = s_dots[wave][lane][c];
      float p2  = ws[OFF_P2N + c];
      float dd  = sqrtf(fmaxf(q2 + p2 - 2.f * dot, 1e-12f));
      z[c] = -dd;
      mx = fmaxf(mx, z[c]);
    }
    float se = 0.f;
#pragma unroll
    for (int c = 0; c < K_CLS; c++) se += expf(z[c] - mx);
    float inv = 1.f / se;
#pragma unroll
    for (int c = 0; c < K_CLS; c++)
      out[(size_t)q * K_CLS + c] = expf(z[c] - mx) * inv;
  }
  if (blockIdx.x == 0 && tid == 0)
    out[(size_t)Q_NUM * K_CLS] = ws[OFF_LOSS] * (1.f / Q_NUM);
}

// ------------------------------------------------------------------
extern "C" void kernel_launch(void* const* d_in, const int* in_sizes, int n_in,
                              void* d_out, int out_size, void* d_ws, size_t ws_size,
                              hipStream_t stream) {
  const float* feat  = (const float*)d_in[0];
  const int*   label = (const int*)d_in[1];
  float* ws  = (float*)d_ws;
  float* out = (float*)d_out;

  hipMemsetAsync(ws + OFF_ZERO, 0, (size_t)ZERO_CNT * sizeof(float), stream);
  k_prep<<<PADC, 256, 0, stream>>>(feat, ws);
  int nBlocks = (Q_NUM + 127) / 128;   // 8 waves/block * 16 queries/wave
  k_pass1<<<nBlocks, 256, 0, stream>>>(feat, label, ws);
  k_stats<<<K_CLS, 256, 0, stream>>>(ws);
  k_select<<<K_CLS, 256, 0, stream>>>(feat, ws);
  k_pass2<<<nBlocks, 256, 0, stream>>>(feat, ws, out);
}